// ReadUnit_1580547966783
// MI455X (gfx1250) — compile-verified
//
#include <hip/hip_runtime.h>
#include <math.h>

typedef __attribute__((ext_vector_type(2))) float v2f;
typedef __attribute__((ext_vector_type(8))) float v8f;
typedef __attribute__((ext_vector_type(4))) unsigned int v4u;
typedef __attribute__((ext_vector_type(8))) int v8i_;
typedef __attribute__((ext_vector_type(4))) int v4i_;

#define DIM     512
#define NTOK    4096
#define BATCH   32
#define TWO_DIM 1024
#define NCHUNK  16            // chunks of N per batch (one block each)
#define TILE_N  64            // tokens per TDM tile
#define NTILES  (NTOK / NCHUNK / TILE_N)   // 4 tiles per block

// ---------------------------------------------------------------------------
// Kernel 1: m_proj[b, j] = sum_k m_prev[b,k] * mem_w[j,k] + mem_b[j]
// One wave per 16x16 output tile; V_WMMA_F32_16X16X4_F32.
// ---------------------------------------------------------------------------
__global__ __launch_bounds__(32) void mproj_kernel(
    const float* __restrict__ m_prev, const float* __restrict__ mem_w,
    const float* __restrict__ mem_b, float* __restrict__ m_proj) {
  const int tile = blockIdx.x;       // 0..63
  const int mt   = tile >> 5;
  const int nt   = tile & 31;
  const int lane = threadIdx.x;
  const int half = lane >> 4;
  const int l15  = lane & 15;
  const int arow = mt * 16 + l15;
  const int ncol = nt * 16 + l15;

  v8f acc = {};
  for (int k0 = 0; k0 < DIM; k0 += 4) {
    const int ka = k0 + (half << 1);
    v2f a, b;
    a[0] = m_prev[arow * DIM + ka];
    a[1] = m_prev[arow * DIM + ka + 1];
    b[0] = mem_w[ncol * DIM + ka];
    b[1] = mem_w[ncol * DIM + ka + 1];
    acc = __builtin_amdgcn_wmma_f32_16x16x4_f32(false, a, false, b,
                                                (short)0, acc, false, false);
  }
  const float bias = mem_b[ncol];
#pragma unroll
  for (int i = 0; i < 8; ++i) {
    const int m = mt * 16 + i + (half << 3);
    m_proj[m * DIM + ncol] = acc[i] + bias;
  }
}

// ---------------------------------------------------------------------------
// Kernel 2: w = c_last*attn_w; u1 = w@concat_w[:,:D]; u2 = w@concat_w[:,D:];
//           v = m_proj*u1 + u2.  Two WMMA accumulators share the A fragment.
// ---------------------------------------------------------------------------
__global__ __launch_bounds__(32) void vvec_kernel(
    const float* __restrict__ c_last, const float* __restrict__ attn_w,
    const float* __restrict__ concat_w, const float* __restrict__ m_proj,
    float* __restrict__ v_out) {
  const int tile = blockIdx.x;
  const int mt   = tile >> 5;
  const int nt   = tile & 31;
  const int lane = threadIdx.x;
  const int half = lane >> 4;
  const int l15  = lane & 15;
  const int arow = mt * 16 + l15;
  const int ncol = nt * 16 + l15;

  v8f acc1 = {}, acc2 = {};
  for (int k0 = 0; k0 < DIM; k0 += 4) {
    const int ka = k0 + (half << 1);
    v2f a, b1, b2;
    a[0] = c_last[arow * DIM + ka]     * attn_w[ka];
    a[1] = c_last[arow * DIM + ka + 1] * attn_w[ka + 1];
    b1[0] = concat_w[ka * TWO_DIM + ncol];
    b1[1] = concat_w[(ka + 1) * TWO_DIM + ncol];
    b2[0] = concat_w[ka * TWO_DIM + DIM + ncol];
    b2[1] = concat_w[(ka + 1) * TWO_DIM + DIM + ncol];
    acc1 = __builtin_amdgcn_wmma_f32_16x16x4_f32(false, a, false, b1,
                                                 (short)0, acc1, false, false);
    acc2 = __builtin_amdgcn_wmma_f32_16x16x4_f32(false, a, false, b2,
                                                 (short)0, acc2, false, false);
  }
#pragma unroll
  for (int i = 0; i < 8; ++i) {
    const int m = mt * 16 + i + (half << 3);
    const float mp = m_proj[m * DIM + ncol];
    v_out[m * DIM + ncol] = mp * acc1[i] + acc2[i];
  }
}

// ---------------------------------------------------------------------------
// TDM: issue a 2D [TILE_N x DIM] tile load k[b, 0:512, n0:n0+64] -> LDS.
// D# built per CDNA5 ISA 8.3/8.4 (group0: count/lds/global/type,
// group1: data_size, dims, tile dims, stride). Groups 2/3 zero (2-D tile).
// This toolchain exposes the 6-arg builtin: (v4u, v8i, v4i, v4i, v8i, cpol).
// ---------------------------------------------------------------------------
__device__ __forceinline__ unsigned lds_off_of(const void* p) {
  return (unsigned)(unsigned long long)(const __attribute__((address_space(3))) void*)p;
}

__device__ __forceinline__ void tdm_load_tile(const float* gptr, unsigned lds_off) {
  const unsigned long long ga = (unsigned long long)gptr;
  v4u g0;
  g0[0] = 1u;                                           // count=1, user desc
  g0[1] = lds_off;                                      // lds_addr (bytes)
  g0[2] = (unsigned)(ga & 0xFFFFFFFFu);                 // global_addr[31:0]
  g0[3] = (unsigned)((ga >> 32) & 0x01FFFFFFu) | (2u << 30);  // addr[56:32]|type=2
  v8i_ g1;
  g1[0] = (int)(2u << 16);        // workgroup_mask=0 | data_size=2 (4B)
  g1[1] = (int)((unsigned)NTOK << 16);   // tensor_dim0[15:0]=4096
  g1[2] = (int)((unsigned)DIM << 16);    // tensor_dim0 hi=0 | tensor_dim1=512
  g1[3] = (int)((unsigned)TILE_N << 16); // tensor_dim1 hi=0 | tile_dim0=64
  g1[4] = (int)DIM;               // tile_dim1=512 | tile_dim2=0
  g1[5] = (int)NTOK;              // tensor_dim0_stride[31:0]=4096
  g1[6] = 0;                      // stride hi | tensor_dim1_stride lo (unused 2D)
  g1[7] = 0;
  const v4i_ z4 = {0, 0, 0, 0};
  const v8i_ z8 = {0, 0, 0, 0, 0, 0, 0, 0};
  __builtin_amdgcn_tensor_load_to_lds(g0, g1, z4, z4, z8, 0);
}

// ---------------------------------------------------------------------------
// Kernel 3 (fused): single streaming pass over k with online softmax.
// Block (chunk p, batch b): 256 tokens as 4 TDM tiles, double-buffered LDS.
// Emits partial (m, Z, racc[512]) per block.
// ---------------------------------------------------------------------------
__global__ __launch_bounds__(256) void attn_fused_kernel(
    const float* __restrict__ kmat, const float* __restrict__ v,
    float* __restrict__ part_r, float* __restrict__ part_m,
    float* __restrict__ part_z) {
  __shared__ float buf[2][DIM * TILE_N];   // 2 x 128 KB
  __shared__ float vs[DIM];
  __shared__ float sred[4 * TILE_N];
  __shared__ float sarr[TILE_N];
  __shared__ float earr[TILE_N];
  __shared__ float bcast[2];               // m_new, scale

  const int p   = blockIdx.x;              // chunk 0..15
  const int b   = blockIdx.y;              // batch 0..31
  const int tid = threadIdx.x;
  const int lane = tid & 31;
  const float* kbase = kmat + (size_t)b * DIM * NTOK + (size_t)p * (NTOK / NCHUNK);

  for (int i = tid; i < DIM; i += 256) vs[i] = v[b * DIM + i];

  float racc0 = 0.f, racc1 = 0.f;          // d = tid, tid+256
  float m_run = -INFINITY, z_run = 0.f;    // meaningful in tid 0 only
  const int d0 = tid, d1 = tid + 256;

  if (tid < 32) tdm_load_tile(kbase, lds_off_of(&buf[0][0]));  // tile 0

  for (int t = 0; t < NTILES; ++t) {
    const float* tb = &buf[t & 1][0];
    if (tid < 32) {
      if (t + 1 < NTILES) {
        tdm_load_tile(kbase + (t + 1) * TILE_N, lds_off_of(&buf[(t + 1) & 1][0]));
        __builtin_amdgcn_s_wait_tensorcnt(1);   // tile t done; t+1 in flight
      } else {
        __builtin_amdgcn_s_wait_tensorcnt(0);
      }
    }
    __syncthreads();                            // buf[t&1] ready (also vs, 1st iter)

    // --- scores: s[n] = sum_d vs[d]*K[d,n], 4-way split over d ---
    {
      const int n = tid & 63, q = tid >> 6;
      float s = 0.f;
#pragma unroll 8
      for (int i = 0; i < 128; ++i) {
        const int d = q * 128 + i;
        s = fmaf(vs[d], tb[d * TILE_N + n], s);
      }
      sred[q * TILE_N + n] = s;
    }
    __syncthreads();
    if (tid < TILE_N)
      sarr[tid] = sred[tid] + sred[TILE_N + tid] + sred[2 * TILE_N + tid] +
                  sred[3 * TILE_N + tid];
    __syncthreads();

    // --- online max update (wave 0) ---
    if (tid < 32) {
      float tm = fmaxf(sarr[tid], sarr[tid + 32]);
      for (int off = 16; off > 0; off >>= 1) tm = fmaxf(tm, __shfl_xor(tm, off, 32));
      if (tid == 0) {
        const float m_new = fmaxf(m_run, tm);
        bcast[0] = m_new;
        bcast[1] = __expf(m_run - m_new);
        m_run = m_new;
      }
    }
    __syncthreads();
    const float m_new = bcast[0];
    const float sc    = bcast[1];
    if (tid < TILE_N) earr[tid] = __expf(sarr[tid] - m_new);
    __syncthreads();

    // --- Z update (wave 0) ---
    if (tid < 32) {
      float zt = earr[tid] + earr[tid + 32];
      for (int off = 16; off > 0; off >>= 1) zt += __shfl_xor(zt, off, 32);
      if (tid == 0) z_run = z_run * sc + zt;
    }

    // --- rescale + accumulate racc (lane-rotated n: conflict-free banks) ---
    racc0 *= sc;
    racc1 *= sc;
#pragma unroll 8
    for (int nn = 0; nn < TILE_N; ++nn) {
      const int n = (nn + lane) & (TILE_N - 1);
      const float e = earr[n];
      racc0 = fmaf(e, tb[d0 * TILE_N + n], racc0);
      racc1 = fmaf(e, tb[d1 * TILE_N + n], racc1);
    }
    __syncthreads();   // done reading buf[t&1] / sred / sarr before reuse
  }

  const int blk = b * NCHUNK + p;
  part_r[(size_t)blk * DIM + d0] = racc0;
  part_r[(size_t)blk * DIM + d1] = racc1;
  if (tid == 0) { part_m[blk] = m_run; part_z[blk] = z_run; }
}

// ---------------------------------------------------------------------------
// Kernel 4: combine partials -> read[b,d]  (1 MB traffic, trivial)
// ---------------------------------------------------------------------------
__global__ __launch_bounds__(256) void combine_kernel(
    const float* __restrict__ part_r, const float* __restrict__ part_m,
    const float* __restrict__ part_z, float* __restrict__ out) {
  const int b = blockIdx.x, tid = threadIdx.x;
  float M = -INFINITY;
#pragma unroll
  for (int p = 0; p < NCHUNK; ++p) M = fmaxf(M, part_m[b * NCHUNK + p]);
  float Z = 0.f;
#pragma unroll
  for (int p = 0; p < NCHUNK; ++p)
    Z += part_z[b * NCHUNK + p] * __expf(part_m[b * NCHUNK + p] - M);
  float a0 = 0.f, a1 = 0.f;
#pragma unroll
  for (int p = 0; p < NCHUNK; ++p) {
    const float w = __expf(part_m[b * NCHUNK + p] - M);
    const size_t base = (size_t)(b * NCHUNK + p) * DIM;
    a0 = fmaf(part_r[base + tid], w, a0);
    a1 = fmaf(part_r[base + tid + 256], w, a1);
  }
  const float inv = 1.0f / Z;
  out[b * DIM + tid]       = a0 * inv;
  out[b * DIM + tid + 256] = a1 * inv;
}

// ---------------------------------------------------------------------------
extern "C" void kernel_launch(void* const* d_in, const int* in_sizes, int n_in,
                              void* d_out, int out_size, void* d_ws, size_t ws_size,
                              hipStream_t stream) {
  const float* memories = (const float*)d_in[0];   // [2, 32, 512]
  const float* kmat     = (const float*)d_in[1];   // [32, 512, 4096]
  const float* c        = (const float*)d_in[2];   // [2, 32, 512]
  const float* mem_w    = (const float*)d_in[3];   // [512, 512]
  const float* mem_b    = (const float*)d_in[4];   // [512]
  const float* concat_w = (const float*)d_in[5];   // [512, 1024]
  // d_in[6]=concat_b, d_in[8]=attn_b: per-batch constants -> cancel in softmax
  const float* attn_w   = (const float*)d_in[7];   // [1, 512]

  const float* m_prev = memories + (size_t)1 * BATCH * DIM;  // memories[-1]
  const float* c_last = c        + (size_t)1 * BATCH * DIM;  // c[-1]

  float* ws     = (float*)d_ws;
  float* m_proj = ws;                                   // 16384 floats
  float* v_vec  = ws + BATCH * DIM;                     // 16384 floats
  float* part_r = ws + 2 * BATCH * DIM;                 // 32*16*512 floats (1 MB)
  float* part_m = part_r + (size_t)BATCH * NCHUNK * DIM;  // 512 floats
  float* part_z = part_m + BATCH * NCHUNK;                // 512 floats
  float* out    = (float*)d_out;                        // [32, 512]

  mproj_kernel<<<64, 32, 0, stream>>>(m_prev, mem_w, mem_b, m_proj);
  vvec_kernel<<<64, 32, 0, stream>>>(c_last, attn_w, concat_w, m_proj, v_vec);
  attn_fused_kernel<<<dim3(NCHUNK, BATCH), 256, 0, stream>>>(
      kmat, v_vec, part_r, part_m, part_z);
  combine_kernel<<<BATCH, 256, 0, stream>>>(part_r, part_m, part_z, out);
}